// GINEModel_17514876633825
// MI455X (gfx1250) — compile-verified
//
#include <hip/hip_runtime.h>
#include <math.h>

// ---------------------------------------------------------------------------
// GINE model for MI455X (gfx1250, wave32).
// Roofline: ~2.6 GFLOP of GEMM (trivial) vs a ~90MB message-passing working
// set that fits the 192MB L2 — *if* the 410MB edge encoding is never
// materialized. So the edge encoder (4x64 + folded BN) is recomputed
// per edge inside the scatter kernel (16B read vs 256B read per edge).
// Dense 64x64 node GEMMs use V_WMMA_F32_16X16X4_F32 (full fp32 path).
// Scatter-add uses inline-asm global_atomic_add_f32 (native L2 atomics,
// no CAS-loop risk) into an L2-resident accumulator.
// ---------------------------------------------------------------------------

typedef __attribute__((ext_vector_type(2))) float v2f;
typedef __attribute__((ext_vector_type(4))) float v4f;
typedef __attribute__((ext_vector_type(8))) float v8f;

#define BN_EPS 1e-5f
#define HDIM 64

// Native CDNA5 f32 atomic add, no-return form (tracked by STOREcnt; kernel
// completion implies S_WAIT_IDLE, and cross-kernel visibility is via L2).
__device__ __forceinline__ void atomic_add_f32(float* addr, float v) {
  asm volatile("global_atomic_add_f32 %0, %1, off"
               :
               : "v"(addr), "v"(v)
               : "memory");
}

__global__ void zero_f32_kernel(float* __restrict__ p, int n) {
  int i = blockIdx.x * blockDim.x + threadIdx.x;
  if (i < n) p[i] = 0.0f;
}

// ---------------------------------------------------------------------------
// Y[r,:] = relu( relu( (X[r,:] (+ AGG[r,:])) @ W + b ) * s + t ),  s/t = folded BN
// X: [nrows, K] row-major, W: [K, 64] row-major, Y: [nrows, 64].
// One wave computes a 16x64 output tile with 4 f32 WMMA accumulators.
// Fragment layouts per CDNA5 ISA 7.12.2 (16x16x4 f32):
//   A 16x4: lanes 0-15 -> M=lane, VGPR{0,1}=K{0,1}; lanes 16-31 -> K{2,3}
//   B 4x16: lanes 0-15 -> N=lane, VGPR{0,1}=K{0,1}; lanes 16-31 -> K{2,3}
//   D 16x16: VGPR r -> row r (lanes 0-15) / row r+8 (lanes 16-31), N=lane&15
// ---------------------------------------------------------------------------
template <int K, bool ADD_AGG>
__global__ __launch_bounds__(256) void block_gemm_wmma_kernel(
    const float* __restrict__ X, const float* __restrict__ AGG,
    const float* __restrict__ W, const float* __restrict__ bias,
    const float* __restrict__ gamma, const float* __restrict__ beta,
    const float* __restrict__ mean, const float* __restrict__ var,
    float* __restrict__ Y, int nrows) {
  const int wave = threadIdx.x >> 5;
  const int lane = threadIdx.x & 31;
  const int tile = blockIdx.x * 8 + wave;   // 16-row tile per wave
  const int r0 = tile << 4;
  if (r0 >= nrows) return;                  // nrows is a multiple of 16 here

  const int half = lane >> 4;               // 0: K pair {0,1}, 1: K pair {2,3}
  const int l = lane & 15;

  v8f acc[4];
#pragma unroll
  for (int t = 0; t < 4; ++t)
#pragma unroll
    for (int r = 0; r < 8; ++r) acc[t][r] = 0.0f;

  const int arow = r0 + l;
#pragma unroll
  for (int kk = 0; kk < K; kk += 4) {
    const int ak = kk + 2 * half;
    v2f a = *(const v2f*)(X + (size_t)arow * K + ak);     // b64 load
    if constexpr (ADD_AGG) {
      const v2f g = *(const v2f*)(AGG + (size_t)arow * HDIM + ak);
      a.x += g.x;
      a.y += g.y;
    }
#pragma unroll
    for (int nt = 0; nt < 4; ++nt) {
      const int n = nt * 16 + l;
      v2f bf;
      bf.x = W[(ak + 0) * HDIM + n];
      bf.y = W[(ak + 1) * HDIM + n];
      acc[nt] = __builtin_amdgcn_wmma_f32_16x16x4_f32(
          false, a, false, bf, (short)0, acc[nt], false, false);
    }
  }

  // Epilogue: bias -> relu -> folded BN -> relu, store.
#pragma unroll
  for (int nt = 0; nt < 4; ++nt) {
    const int n = nt * 16 + l;
    const float s = gamma[n] * rsqrtf(var[n] + BN_EPS);
    const float t = beta[n] - mean[n] * s;
    const float bn = bias[n];
#pragma unroll
    for (int r = 0; r < 8; ++r) {
      const int row = r0 + r + 8 * half;
      float y = acc[nt][r] + bn;
      y = fmaxf(y, 0.0f);
      y = fmaxf(y * s + t, 0.0f);
      Y[(size_t)row * HDIM + n] = y;
    }
  }
}

// ---------------------------------------------------------------------------
// Fused: edge encode (recomputed, BN folded) + gather h[src] + relu + scatter
// atomic-add into agg[dst]. One wave per edge (2 features per lane, so src/dst
// and edge_attr loads are wave-uniform -> scalarizable).
// ---------------------------------------------------------------------------
__global__ __launch_bounds__(256) void edge_scatter_kernel(
    const int* __restrict__ ei, const float* __restrict__ ea,
    const float* __restrict__ h, float* __restrict__ agg,
    const float* __restrict__ eW, const float* __restrict__ eb,
    const float* __restrict__ eg, const float* __restrict__ ebeta,
    const float* __restrict__ emean, const float* __restrict__ evar,
    long long E) {
  const long long gtid = (long long)blockIdx.x * blockDim.x + threadIdx.x;
  const long long edge = gtid >> 5;
  if (edge >= E) return;
  const int f = (int)(gtid & 31) * 2;

  // Prefetch the streamed edge_attr a few waves ahead (global_prefetch_b8).
  if (edge + 512 < E) __builtin_prefetch(&ea[(edge + 512) * 4], 0, 1);

  const int src = ei[edge];
  const int dst = ei[E + edge];
  const v4f a = *(const v4f*)(ea + edge * 4);                    // b128 load
  const v2f h2 = *(const v2f*)(h + (size_t)src * HDIM + f);      // b64 gather

  float msg[2];
#pragma unroll
  for (int j = 0; j < 2; ++j) {
    const int n = f + j;
    float y = a.x * eW[0 * HDIM + n] + a.y * eW[1 * HDIM + n] +
              a.z * eW[2 * HDIM + n] + a.w * eW[3 * HDIM + n] + eb[n];
    y = fmaxf(y, 0.0f);
    const float s = eg[n] * rsqrtf(evar[n] + BN_EPS);
    const float t = ebeta[n] - emean[n] * s;
    const float e = fmaxf(y * s + t, 0.0f);
    msg[j] = fmaxf(h2[j] + e, 0.0f);
  }
  float* dstp = &agg[(size_t)dst * HDIM + f];
  atomic_add_f32(dstp + 0, msg[0]);
  atomic_add_f32(dstp + 1, msg[1]);
}

// pooled[batch[node], f] += h[node, f]   (pooled = 512KB, L2-resident)
__global__ __launch_bounds__(256) void pool_kernel(
    const float* __restrict__ h, const int* __restrict__ batch,
    float* __restrict__ pooled, int total) {
  const int i = blockIdx.x * blockDim.x + threadIdx.x;
  if (i >= total) return;
  const int node = i >> 6;
  const int f = i & 63;
  atomic_add_f32(&pooled[(size_t)batch[node] * HDIM + f], h[i]);
}

// Per-graph head: desc block (12->64) + concat(pooled,d) -> 128->64 relu -> 64->1 sigmoid
__global__ __launch_bounds__(64) void final_kernel(
    const float* __restrict__ pooled, const float* __restrict__ desc,
    const float* __restrict__ dW, const float* __restrict__ db,
    const float* __restrict__ dg, const float* __restrict__ dbeta,
    const float* __restrict__ dmean, const float* __restrict__ dvar,
    const float* __restrict__ W1, const float* __restrict__ b1,
    const float* __restrict__ W2, const float* __restrict__ b2,
    float* __restrict__ out) {
  __shared__ float c[2 * HDIM];
  __shared__ float mid[HDIM];
  const int g = blockIdx.x;
  const int f = threadIdx.x;  // 0..63

  float y = db[f];
#pragma unroll
  for (int k = 0; k < 12; ++k) y += desc[g * 12 + k] * dW[k * HDIM + f];
  y = fmaxf(y, 0.0f);
  const float s = dg[f] * rsqrtf(dvar[f] + BN_EPS);
  const float t = dbeta[f] - dmean[f] * s;
  c[f] = pooled[(size_t)g * HDIM + f];
  c[HDIM + f] = fmaxf(y * s + t, 0.0f);
  __syncthreads();

  float m = b1[f];
#pragma unroll
  for (int k = 0; k < 2 * HDIM; ++k) m += c[k] * W1[k * HDIM + f];
  mid[f] = fmaxf(m, 0.0f);
  __syncthreads();

  if (f == 0) {
    float o = b2[0];
#pragma unroll
    for (int k = 0; k < HDIM; ++k) o += mid[k] * W2[k];
    out[g] = 1.0f / (1.0f + expf(-o));
  }
}

// ---------------------------------------------------------------------------
// Host launcher.
// Input order: x, edge_index, edge_attr, batch, descriptors, then params as
// JAX pytree leaves (sorted keys): convs[0..2]{b,beta,gamma,mean,var,w},
// desc{...}, edge_enc{...}, final_b1, final_b2, final_w1, final_w2,
// node_enc{b,beta,gamma,mean,var,w}.
// ---------------------------------------------------------------------------
extern "C" void kernel_launch(void* const* d_in, const int* in_sizes, int n_in,
                              void* d_out, int out_size, void* d_ws, size_t ws_size,
                              hipStream_t stream) {
  const float* x = (const float*)d_in[0];
  const int* edge_index = (const int*)d_in[1];
  const float* edge_attr = (const float*)d_in[2];
  const int* batch = (const int*)d_in[3];
  const float* descriptors = (const float*)d_in[4];

  const int N = in_sizes[0] / 8;
  const long long E = in_sizes[2] / 4;
  const int G = in_sizes[4] / 12;

  // params: blocks have leaf order b,beta,gamma,mean,var,w
  const float* const* P = (const float* const*)d_in;
  const int CONV0 = 5;           // convs[0..2]: 5..22
  const int DESC = 23;           // 23..28
  const int EDGE = 29;           // 29..34
  const float* final_b1 = P[35];
  const float* final_b2 = P[36];
  const float* final_w1 = P[37];
  const float* final_w2 = P[38];
  const int NODE = 39;           // 39..44

  // workspace carve: h | agg | pooled  (~52MB)
  float* h = (float*)d_ws;
  float* agg = h + (size_t)N * HDIM;
  float* pooled = agg + (size_t)N * HDIM;

  const int tiles = (N + 15) / 16;
  const int gemmBlocks = (tiles + 7) / 8;
  const int nh = N * HDIM;

  // 1) node encoder: h = block(x, node_enc), K = 8
  block_gemm_wmma_kernel<8, false><<<gemmBlocks, 256, 0, stream>>>(
      x, nullptr, P[NODE + 5], P[NODE + 0], P[NODE + 2], P[NODE + 1],
      P[NODE + 3], P[NODE + 4], h, N);

  // 2) three GINE conv layers
  const long long etotal = E * 32;
  const int eblocks = (int)((etotal + 255) / 256);
  for (int layer = 0; layer < 3; ++layer) {
    const int C = CONV0 + layer * 6;
    zero_f32_kernel<<<(nh + 255) / 256, 256, 0, stream>>>(agg, nh);
    edge_scatter_kernel<<<eblocks, 256, 0, stream>>>(
        edge_index, edge_attr, h, agg,
        P[EDGE + 5], P[EDGE + 0], P[EDGE + 2], P[EDGE + 1], P[EDGE + 3],
        P[EDGE + 4], E);
    block_gemm_wmma_kernel<64, true><<<gemmBlocks, 256, 0, stream>>>(
        h, agg, P[C + 5], P[C + 0], P[C + 2], P[C + 1], P[C + 3], P[C + 4],
        h, N);  // in-place: each wave reads only the 16 rows it writes
  }

  // 3) global add pool
  const int pg = G * HDIM;
  zero_f32_kernel<<<(pg + 255) / 256, 256, 0, stream>>>(pooled, pg);
  pool_kernel<<<(nh + 255) / 256, 256, 0, stream>>>(h, batch, pooled, nh);

  // 4) head
  final_kernel<<<G, 64, 0, stream>>>(
      pooled, descriptors, P[DESC + 5], P[DESC + 0], P[DESC + 2], P[DESC + 1],
      P[DESC + 3], P[DESC + 4], final_w1, final_b1, final_w2, final_b2,
      (float*)d_out);
}